// TopKActivation_49795850829959
// MI455X (gfx1250) — compile-verified
//
#include <hip/hip_runtime.h>
#include <stdint.h>

// Top-k (k=64) per row of an 8192 x 24576 fp32 matrix: keep top-k values at
// their positions, zero elsewhere. Memory-bound: exactly 1 global read +
// 1 global write per element. Row staged to LDS via CDNA5 async
// global->LDS b128 copies; k-th-largest found by 4-pass MSB radix select
// over order-preserving u32 keys using LDS histogram atomics.

#define COLS 24576
#define NTHR 256

__device__ __forceinline__ uint32_t map_key(float f) {
    // Monotone map: float total order -> unsigned integer order.
    uint32_t b = __float_as_uint(f);
    return b ^ (uint32_t)(((int32_t)b >> 31) | 0x80000000);
}

__global__ __launch_bounds__(NTHR) void topk_row_kernel(
    const float* __restrict__ z, const int* __restrict__ kptr,
    float* __restrict__ out) {
    __shared__ float    srow[COLS];   // 96 KB row stage
    __shared__ uint32_t bins[256];    // radix histogram
    __shared__ uint32_t s_sel;        // chosen radix digit this pass
    __shared__ uint32_t s_kk;         // remaining rank within chosen bin
    __shared__ uint32_t s_tie;        // tie-break counter at threshold

    const int tid = threadIdx.x;
    const int row = blockIdx.x;
    const uint32_t k = (uint32_t)(*kptr);

    const float* gsrc = z + (size_t)row * COLS;
    float*       gdst = out + (size_t)row * COLS;

    // ---- Stage row into LDS with async global->LDS copies (ASYNCcnt) ----
    // Generic address of an LDS object carries the LDS byte offset in its
    // low 32 bits (aperture in [63:32]), which is what the async-load VDST
    // VGPR expects (hardware adds LDS_BASE).
    const uint32_t lds_base = (uint32_t)(uintptr_t)&srow[0];
    const int nvec = COLS / 4;  // 6144 float4 tiles
    for (int v = tid; v < nvec; v += NTHR) {
        uint64_t ga = (uint64_t)(uintptr_t)(gsrc + v * 4);
        uint32_t la = lds_base + (uint32_t)v * 16u;
        asm volatile("global_load_async_to_lds_b128 %0, %1, off"
                     :: "v"(la), "v"(ga)
                     : "memory");
    }
    asm volatile("s_wait_asynccnt 0" ::: "memory");
    __syncthreads();

    // ---- 4-pass MSB-first radix select of the k-th largest key ----
    const float4* srow4 = reinterpret_cast<const float4*>(srow);
    uint32_t prefix = 0, pmask = 0;
    uint32_t kk = k;  // rank (1-based, from the top) within candidate set

    for (int pass = 0; pass < 4; ++pass) {
        const int shift = 24 - 8 * pass;
        bins[tid & 255] = 0;
        __syncthreads();

        for (int v = tid; v < nvec; v += NTHR) {
            float4 f = srow4[v];
            uint32_t kx = map_key(f.x), ky = map_key(f.y);
            uint32_t kz = map_key(f.z), kw = map_key(f.w);
            if ((kx & pmask) == prefix) atomicAdd(&bins[(kx >> shift) & 0xFF], 1u);
            if ((ky & pmask) == prefix) atomicAdd(&bins[(ky >> shift) & 0xFF], 1u);
            if ((kz & pmask) == prefix) atomicAdd(&bins[(kz >> shift) & 0xFF], 1u);
            if ((kw & pmask) == prefix) atomicAdd(&bins[(kw >> shift) & 0xFF], 1u);
        }
        __syncthreads();

        if (tid == 0) {
            uint32_t cum = 0;
            uint32_t sel = 0, nkk = 1;
            for (int b = 255; b >= 0; --b) {
                uint32_t c = bins[b];
                if (cum + c >= kk) { sel = (uint32_t)b; nkk = kk - cum; break; }
                cum += c;
            }
            s_sel = sel;
            s_kk  = nkk;
        }
        __syncthreads();

        prefix |= (s_sel << shift);
        pmask  |= (0xFFu << shift);
        kk = s_kk;
        // next pass's leading barrier orders the s_sel/s_kk reads vs rewrite
    }

    const uint32_t tkey  = prefix;  // key of the k-th largest element
    const uint32_t allow = kk;      // how many ties at tkey to keep
    if (tid == 0) s_tie = 0;
    __syncthreads();

    // ---- Output pass: keep top-k in place, zero elsewhere ----
    for (int v = tid; v < nvec; v += NTHR) {
        float4 f = srow4[v];
        float in[4] = {f.x, f.y, f.z, f.w};
        float o[4];
#pragma unroll
        for (int j = 0; j < 4; ++j) {
            uint32_t kj = map_key(in[j]);
            float r = 0.0f;
            if (kj > tkey) {
                r = in[j];
            } else if (kj == tkey) {
                if (atomicAdd(&s_tie, 1u) < allow) r = in[j];
            }
            o[j] = r;
        }
        reinterpret_cast<float4*>(gdst)[v] = make_float4(o[0], o[1], o[2], o[3]);
    }
}

extern "C" void kernel_launch(void* const* d_in, const int* in_sizes, int n_in,
                              void* d_out, int out_size, void* d_ws, size_t ws_size,
                              hipStream_t stream) {
    (void)n_in; (void)d_ws; (void)ws_size; (void)out_size;
    const float* z    = (const float*)d_in[0];
    const int*   kptr = (const int*)d_in[1];
    float*       out  = (float*)d_out;

    const int rows = in_sizes[0] / COLS;  // 8192
    topk_row_kernel<<<dim3(rows), dim3(NTHR), 0, stream>>>(z, kptr, out);
}